// Non_Local_20736102105443
// MI455X (gfx1250) — compile-verified
//
#include <hip/hip_runtime.h>

// ---------------------------------------------------------------------------
// Non-local block, algebraically collapsed:
//   phi = phi_w @ X + phi_b          [128 x 4096]  per batch   (X = x[b] as 256x4096)
//   g   = g_w   @ X + g_b            [128 x 4096]
//   M   = g @ phi^T                  [128 x 128]   (K = 4096, split 16 ways)
//   W2  = (w_w @ M) / N              [256 x 128]
//   z   = W2 @ phi + w_b + x         [256 x 4096]
// Everything runs on V_WMMA_F32_16X16X4_F32 (exact fp32, wave32 matrix pipe).
// ---------------------------------------------------------------------------

typedef float v2f __attribute__((ext_vector_type(2)));
typedef float v8f __attribute__((ext_vector_type(8)));

static constexpr int NB     = 4;     // batches
static constexpr int CIN    = 256;   // input channels
static constexpr int CI     = 128;   // inter channels
static constexpr int NPIX   = 4096;  // H*W
static constexpr int KSPLIT = 16;    // K-split for the Gram stage

#define WMMA_F32(a, b, c) \
  __builtin_amdgcn_wmma_f32_16x16x4_f32(false, (a), false, (b), (short)0, (c), false, false)

// Per-wave 32(M) x 64(N) output tile. A row-major (contiguous along K).
// BTRANS=false: B(k,n) = B[k*ldb + n]   (normal row-major K x N)
// BTRANS=true : B(k,n) = B[n*ldb + k]   (i.e. computes A @ B^T, both K-major)
template <bool BTRANS, bool HAS_BIAS, bool HAS_RES>
__device__ __forceinline__ void wave_gemm(
    const float* __restrict__ A, int lda,
    const float* __restrict__ B, int ldb,
    float* __restrict__ C, int ldc,
    const float* __restrict__ bias,
    const float* __restrict__ res, int ldr,
    int K, float alpha, int row0, int col0)
{
  const int lane = (int)(threadIdx.x & 31u);
  const int r    = lane & 15;   // row (A) / col (B) within 16
  const int h    = lane >> 4;   // K-pair selector: lanes 16..31 carry K+2,K+3

  v8f acc[2][4];
#pragma unroll
  for (int i = 0; i < 2; ++i)
#pragma unroll
    for (int j = 0; j < 4; ++j)
      acc[i][j] = v8f{};

  for (int k = 0; k < K; k += 4) {
    // A fragment: 16x4 f32 -> lane r holds A[row][k+2h], A[row][k+2h+1]
    v2f a[2];
#pragma unroll
    for (int i = 0; i < 2; ++i) {
      const float* ap = A + (size_t)(row0 + i * 16 + r) * lda + (k + 2 * h);
      a[i] = *(const v2f*)ap;   // 8B aligned: global_load_b64
    }
    // B fragment: 4x16 f32 -> lane r holds B[k+2h][col], B[k+2h+1][col]
    v2f b[4];
#pragma unroll
    for (int j = 0; j < 4; ++j) {
      const int c = col0 + j * 16 + r;
      if (BTRANS) {
        const float* bp = B + (size_t)c * ldb + (k + 2 * h);
        b[j] = *(const v2f*)bp; // contiguous along K for the NT (Gram) case
      } else {
        const int kk = k + 2 * h;
        b[j].x = B[(size_t)kk * ldb + c];
        b[j].y = B[(size_t)(kk + 1) * ldb + c];
      }
    }
#pragma unroll
    for (int i = 0; i < 2; ++i)
#pragma unroll
      for (int j = 0; j < 4; ++j)
        acc[i][j] = WMMA_F32(a[i], b[j], acc[i][j]);
  }

  // Epilogue: C/D layout -> VGPR e: lanes 0-15 = row e, lanes 16-31 = row e+8
#pragma unroll
  for (int i = 0; i < 2; ++i) {
#pragma unroll
    for (int j = 0; j < 4; ++j) {
      const int col = col0 + j * 16 + r;
#pragma unroll
      for (int e = 0; e < 8; ++e) {
        const int row = row0 + i * 16 + 8 * h + e;
        float v = acc[i][j][e] * alpha;
        if (HAS_BIAS) v += bias[row];
        if (HAS_RES)  v += res[(size_t)row * ldr + col];
        C[(size_t)row * ldc + col] = v;
      }
    }
  }
}

// ---- stage 1: phi = phi_w@X + phi_b ; g = g_w@X + g_b  -------------------
__global__ __launch_bounds__(256) void k_conv_in(
    const float* __restrict__ x,
    const float* __restrict__ phi_w, const float* __restrict__ phi_b,
    const float* __restrict__ g_w,   const float* __restrict__ g_b,
    float* __restrict__ phi, float* __restrict__ g)
{
  int wid = (int)((blockIdx.x * blockDim.x + threadIdx.x) >> 5);
  const int nt = wid & 63; wid >>= 6;          // 64 N-tiles of 64
  const int mt = wid & 3;  wid >>= 2;          // 4 M-tiles of 32
  const int b  = wid & 3;  wid >>= 2;          // 4 batches
  const int ws = wid;                           // 0=phi, 1=g
  if (ws >= 2) return;                          // wave-uniform guard

  const float* A    = ws ? g_w : phi_w;
  const float* bias = ws ? g_b : phi_b;
  float* out        = (ws ? g : phi) + (size_t)b * CI * NPIX;
  const float* Bm   = x + (size_t)b * CIN * NPIX;

  wave_gemm<false, true, false>(A, CIN, Bm, NPIX, out, NPIX,
                                bias, nullptr, 0,
                                CIN, 1.0f, mt * 32, nt * 64);
}

// ---- stage 2: Mpart[b][s] = g[:, sK:(s+1)K] @ phi[:, sK:(s+1)K]^T --------
__global__ __launch_bounds__(256) void k_gram(
    const float* __restrict__ phi, const float* __restrict__ g,
    float* __restrict__ Mpart)
{
  int wid = (int)((blockIdx.x * blockDim.x + threadIdx.x) >> 5);
  const int ntt = wid & 1;  wid >>= 1;         // 2 N-tiles of 64
  const int mt  = wid & 3;  wid >>= 2;         // 4 M-tiles of 32
  const int sp  = wid & 15; wid >>= 4;         // 16 K-chunks
  const int b   = wid;
  if (b >= NB) return;

  const int kchunk = NPIX / KSPLIT;            // 256
  const float* A  = g   + (size_t)b * CI * NPIX + (size_t)sp * kchunk;
  const float* Bm = phi + (size_t)b * CI * NPIX + (size_t)sp * kchunk;
  float* C = Mpart + ((size_t)b * KSPLIT + sp) * CI * CI;

  wave_gemm<true, false, false>(A, NPIX, Bm, NPIX, C, CI,
                                nullptr, nullptr, 0,
                                kchunk, 1.0f, mt * 32, ntt * 64);
}

// ---- stage 3: deterministic K-split reduction ----------------------------
__global__ __launch_bounds__(256) void k_reduce(
    const float* __restrict__ Mpart, float* __restrict__ M)
{
  const int idx = (int)(blockIdx.x * blockDim.x + threadIdx.x);
  if (idx >= NB * CI * CI) return;
  const int b = idx / (CI * CI);
  const int e = idx % (CI * CI);
  const float* p = Mpart + (size_t)b * KSPLIT * CI * CI + e;
  float s = 0.0f;
#pragma unroll
  for (int sp = 0; sp < KSPLIT; ++sp) s += p[(size_t)sp * CI * CI];
  M[idx] = s;
}

// ---- stage 4: W2 = (w_w @ M) / N -----------------------------------------
__global__ __launch_bounds__(256) void k_w2(
    const float* __restrict__ w_w, const float* __restrict__ M,
    float* __restrict__ W2)
{
  int wid = (int)((blockIdx.x * blockDim.x + threadIdx.x) >> 5);
  const int ntt = wid & 1; wid >>= 1;          // 2 N-tiles of 64
  const int mt  = wid & 7; wid >>= 3;          // 8 M-tiles of 32
  const int b   = wid;
  if (b >= NB) return;

  wave_gemm<false, false, false>(w_w, CI, M + (size_t)b * CI * CI, CI,
                                 W2 + (size_t)b * CIN * CI, CI,
                                 nullptr, nullptr, 0,
                                 CI, 1.0f / (float)NPIX, mt * 32, ntt * 64);
}

// ---- stage 5: z = W2 @ phi + w_b + x -------------------------------------
__global__ __launch_bounds__(256) void k_out(
    const float* __restrict__ W2, const float* __restrict__ phi,
    const float* __restrict__ w_b, const float* __restrict__ x,
    float* __restrict__ z)
{
  int wid = (int)((blockIdx.x * blockDim.x + threadIdx.x) >> 5);
  const int nt = wid & 63; wid >>= 6;          // 64 N-tiles of 64
  const int mt = wid & 7;  wid >>= 3;          // 8 M-tiles of 32
  const int b  = wid;
  if (b >= NB) return;

  wave_gemm<false, true, true>(W2 + (size_t)b * CIN * CI, CI,
                               phi + (size_t)b * CI * NPIX, NPIX,
                               z + (size_t)b * CIN * NPIX, NPIX,
                               w_b,
                               x + (size_t)b * CIN * NPIX, NPIX,
                               CI, 1.0f, mt * 32, nt * 64);
}

// ---------------------------------------------------------------------------
extern "C" void kernel_launch(void* const* d_in, const int* in_sizes, int n_in,
                              void* d_out, int out_size, void* d_ws, size_t ws_size,
                              hipStream_t stream)
{
  const float* x     = (const float*)d_in[0];
  const float* g_w   = (const float*)d_in[1];
  const float* g_b   = (const float*)d_in[2];
  // d_in[3], d_in[4] = theta_w/theta_b: dead code in the reference, skipped.
  const float* phi_w = (const float*)d_in[5];
  const float* phi_b = (const float*)d_in[6];
  const float* w_w   = (const float*)d_in[7];
  const float* w_b   = (const float*)d_in[8];
  float* z = (float*)d_out;

  // workspace layout (floats): phi | g | Mpart | M | W2   (~21.8 MB total)
  float* phi   = (float*)d_ws;
  float* g     = phi   + (size_t)NB * CI * NPIX;
  float* Mpart = g     + (size_t)NB * CI * NPIX;
  float* M     = Mpart + (size_t)NB * KSPLIT * CI * CI;
  float* W2    = M     + (size_t)NB * CI * CI;

  const dim3 blk(256);
  // waves: 2*4*4*64 = 2048 -> 256 blocks of 8 waves
  k_conv_in<<<256, blk, 0, stream>>>(x, phi_w, phi_b, g_w, g_b, phi, g);
  // waves: 4*16*4*2 = 512 -> 64 blocks
  k_gram<<<64, blk, 0, stream>>>(phi, g, Mpart);
  // 65536 scalar outputs
  k_reduce<<<(NB * CI * CI + 255) / 256, blk, 0, stream>>>(Mpart, M);
  // waves: 4*8*2 = 64 -> 8 blocks
  k_w2<<<8, blk, 0, stream>>>(w_w, M, W2);
  // waves: 4*8*64 = 2048 -> 256 blocks
  k_out<<<256, blk, 0, stream>>>(W2, phi, w_b, x, z);
}